// GCN_67765993996292
// MI455X (gfx1250) — compile-verified
//
#include <hip/hip_runtime.h>

#define NN 100000
#define EE 1600000
#define FHID 128
#define FOUT 64
#define BN_ROWS 400

typedef float v2f __attribute__((ext_vector_type(2)));
typedef float v8f __attribute__((ext_vector_type(8)));

// -------------------- degree / norm --------------------
__global__ void k_deg_init(float* deg, int n) {
    int i = blockIdx.x * blockDim.x + threadIdx.x;
    if (i < n) deg[i] = 1.0f;  // self-loop
}

__global__ void k_deg_edge(const int* __restrict__ dst, float* deg, int e) {
    int i = blockIdx.x * blockDim.x + threadIdx.x;
    if (i < e) atomicAdd(&deg[dst[i]], 1.0f);
}

__global__ void k_deg_fin(float* deg, int n) {
    int i = blockIdx.x * blockDim.x + threadIdx.x;
    if (i < n) deg[i] = rsqrtf(deg[i]);   // deg >= 1 always
}

// -------------------- fp32 WMMA GEMM: H[N,F] = X[N,128] @ W[128,F] --------------------
// one block = 16 output rows, 128 threads = 4 waves; wave w owns col-tiles w, w+4, ...
template <int F>
__global__ __launch_bounds__(128) void k_gemm_wmma(const float* __restrict__ X,
                                                   const float* __restrict__ W,
                                                   float* __restrict__ H) {
    constexpr int K = 128;
    __shared__ float wlds[K * F];     // full weight matrix (<= 64 KB, LDS is 320 KB/WGP)
    __shared__ float xlds[16 * K];    // 16-row input tile (8 KB)

    const int tid  = threadIdx.x;
    const int row0 = blockIdx.x * 16;

    for (int i = tid; i < K * F; i += 128) wlds[i] = W[i];
    for (int i = tid; i < 16 * K; i += 128) {
        int r = i >> 7, c = i & 127;
        xlds[i] = X[(size_t)(row0 + r) * K + c];
    }
    __syncthreads();

    const int wave = tid >> 5;          // wave32
    const int lane = tid & 31;
    const int m    = lane & 15;         // row (A) / col (B,D) within tile
    const int ksel = (lane >> 4) * 2;   // lanes 16-31 hold K+2,K+3

    for (int ct = wave; ct * 16 < F; ct += 4) {
        const int col0 = ct * 16;
        v8f acc = {0.f, 0.f, 0.f, 0.f, 0.f, 0.f, 0.f, 0.f};
        #pragma unroll
        for (int k0 = 0; k0 < K; k0 += 4) {
            v2f a, b;
            a.x = xlds[m * K + k0 + ksel];
            a.y = xlds[m * K + k0 + ksel + 1];
            b.x = wlds[(k0 + ksel) * F + col0 + m];
            b.y = wlds[(k0 + ksel + 1) * F + col0 + m];
            acc = __builtin_amdgcn_wmma_f32_16x16x4_f32(
                false, a, false, b, (short)0, acc, false, false);
        }
        const int rbase = (lane >> 4) * 8;   // VGPR i of C/D = row rbase+i
        #pragma unroll
        for (int i = 0; i < 8; ++i)
            H[(size_t)(row0 + rbase + i) * F + col0 + m] = acc[i];
    }
}

// -------------------- aggregation --------------------
// out[v,f] = bias[f] + dinv[v]^2 * h[v,f]   (self-loop term + bias)
template <int F>
__global__ void k_agg_init(const float* __restrict__ h, const float* __restrict__ bias,
                           const float* __restrict__ dinv, float* __restrict__ out) {
    int idx = blockIdx.x * blockDim.x + threadIdx.x;
    if (idx >= NN * F) return;
    int v = idx / F, f = idx % F;
    float di = dinv[v];
    out[idx] = bias[f] + di * di * h[idx];
}

// one wave per edge; lane handles float4 chunks (F=128 -> exactly 32 lanes * 4)
template <int F>
__global__ __launch_bounds__(256) void k_agg_edge(const float* __restrict__ h,
                                                  const int* __restrict__ src,
                                                  const int* __restrict__ dst,
                                                  const float* __restrict__ dinv,
                                                  float* __restrict__ out) {
    int gid  = blockIdx.x * blockDim.x + threadIdx.x;
    int e    = gid >> 5;
    int lane = gid & 31;
    if (e >= EE) return;
    int s = src[e], d = dst[e];
    float norm = dinv[s] * dinv[d];
    const float* hs = h + (size_t)s * F;
    float* od = out + (size_t)d * F;
    for (int f = lane * 4; f < F; f += 128) {
        float4 v = *reinterpret_cast<const float4*>(hs + f);
        atomicAdd(od + f + 0, v.x * norm);
        atomicAdd(od + f + 1, v.y * norm);
        atomicAdd(od + f + 2, v.z * norm);
        atomicAdd(od + f + 3, v.w * norm);
    }
}

// -------------------- batch norm --------------------
__global__ void k_stats_zero(float* stats) {
    int i = blockIdx.x * blockDim.x + threadIdx.x;
    if (i < 256) stats[i] = 0.0f;
}

__global__ __launch_bounds__(256) void k_bn_stats(const float* __restrict__ G,
                                                  float* __restrict__ sum,
                                                  float* __restrict__ sumsq) {
    __shared__ float ls[256], ls2[256];
    int f    = threadIdx.x & 127;
    int half = threadIdx.x >> 7;
    int r0   = blockIdx.x * BN_ROWS;
    int rend = r0 + BN_ROWS; if (rend > NN) rend = NN;
    float s = 0.f, s2 = 0.f;
    for (int r = r0 + half; r < rend; r += 2) {
        float v = G[(size_t)r * 128 + f];
        s += v; s2 += v * v;
    }
    ls[threadIdx.x] = s; ls2[threadIdx.x] = s2;
    __syncthreads();
    if (half == 0) {
        atomicAdd(&sum[f],   s  + ls [threadIdx.x + 128]);
        atomicAdd(&sumsq[f], s2 + ls2[threadIdx.x + 128]);
    }
}

// h_out = (G-m)*gamma*rsqrt(var+eps)+beta + res ; act_out = relu(h_out)
__global__ void k_bn_apply(const float* __restrict__ G, const float* __restrict__ res,
                           const float* __restrict__ gamma, const float* __restrict__ beta,
                           const float* __restrict__ sum, const float* __restrict__ sumsq,
                           float* __restrict__ h_out, float* __restrict__ act_out) {
    int idx = blockIdx.x * blockDim.x + threadIdx.x;
    if (idx >= NN * 128) return;
    int f = idx & 127;
    const float inv_n = 1.0f / (float)NN;
    float m   = sum[f] * inv_n;
    float var = sumsq[f] * inv_n - m * m;
    float sc  = gamma[f] * rsqrtf(var + 1e-5f);
    float v   = (G[idx] - m) * sc + beta[f] + res[idx];
    h_out[idx]   = v;
    act_out[idx] = v > 0.f ? v : 0.f;
}

// -------------------- launcher --------------------
extern "C" void kernel_launch(void* const* d_in, const int* in_sizes, int n_in,
                              void* d_out, int out_size, void* d_ws, size_t ws_size,
                              hipStream_t stream) {
    const float* x   = (const float*)d_in[0];
    const int*   ei  = (const int*)d_in[1];          // [2, E]: row 0 = src, row 1 = dst
    const int*   src = ei;
    const int*   dst = ei + EE;
    const float* W1  = (const float*)d_in[2];
    const float* b1  = (const float*)d_in[3];
    const float* g1  = (const float*)d_in[4];
    const float* be1 = (const float*)d_in[5];
    const float* W2  = (const float*)d_in[6];
    const float* b2  = (const float*)d_in[7];
    const float* g2  = (const float*)d_in[8];
    const float* be2 = (const float*)d_in[9];
    const float* W3  = (const float*)d_in[10];
    const float* b3  = (const float*)d_in[11];
    float* out = (float*)d_out;

    float* ws    = (float*)d_ws;
    float* dinv  = ws;                       // N
    float* Hbuf  = dinv + NN;                // N*128
    float* Gbuf  = Hbuf + (size_t)NN * 128;  // N*128
    float* h1    = Gbuf + (size_t)NN * 128;  // N*128 residual
    float* act   = h1   + (size_t)NN * 128;  // N*128 relu activation
    float* stats = act  + (size_t)NN * 128;  // 256 (sum | sumsq)
    float* bsum  = stats;
    float* bsq   = stats + 128;

    const int TPB = 256;
    dim3 blkN((NN + TPB - 1) / TPB), tpb(TPB);
    dim3 blkE((EE + TPB - 1) / TPB);
    dim3 blkNF128(((size_t)NN * 128 + TPB - 1) / TPB);
    dim3 blkNF64(((size_t)NN * 64 + TPB - 1) / TPB);
    dim3 blkEdgeWave(((size_t)EE * 32 + TPB - 1) / TPB);
    dim3 blkGemm(NN / 16), tgemm(128);
    dim3 blkBN((NN + BN_ROWS - 1) / BN_ROWS);

    // degrees -> dinv (in place)
    k_deg_init<<<blkN, tpb, 0, stream>>>(dinv, NN);
    k_deg_edge<<<blkE, tpb, 0, stream>>>(dst, dinv, EE);
    k_deg_fin <<<blkN, tpb, 0, stream>>>(dinv, NN);

    // ---- layer 1 ----
    k_gemm_wmma<128><<<blkGemm, tgemm, 0, stream>>>(x, W1, Hbuf);
    k_agg_init<128><<<blkNF128, tpb, 0, stream>>>(Hbuf, b1, dinv, Gbuf);
    k_agg_edge<128><<<blkEdgeWave, tpb, 0, stream>>>(Hbuf, src, dst, dinv, Gbuf);
    k_stats_zero<<<1, 256, 0, stream>>>(stats);
    k_bn_stats<<<blkBN, tpb, 0, stream>>>(Gbuf, bsum, bsq);
    k_bn_apply<<<blkNF128, tpb, 0, stream>>>(Gbuf, x, g1, be1, bsum, bsq, h1, act);

    // ---- layer 2 ----
    k_gemm_wmma<128><<<blkGemm, tgemm, 0, stream>>>(act, W2, Hbuf);
    k_agg_init<128><<<blkNF128, tpb, 0, stream>>>(Hbuf, b2, dinv, Gbuf);
    k_agg_edge<128><<<blkEdgeWave, tpb, 0, stream>>>(Hbuf, src, dst, dinv, Gbuf);
    k_stats_zero<<<1, 256, 0, stream>>>(stats);
    k_bn_stats<<<blkBN, tpb, 0, stream>>>(Gbuf, bsum, bsq);
    k_bn_apply<<<blkNF128, tpb, 0, stream>>>(Gbuf, h1, g2, be2, bsum, bsq, h1, act);

    // ---- layer 3 (no bn/skip/relu) ----
    k_gemm_wmma<64><<<blkGemm, tgemm, 0, stream>>>(act, W3, Hbuf);
    k_agg_init<64><<<blkNF64, tpb, 0, stream>>>(Hbuf, b3, dinv, out);
    k_agg_edge<64><<<blkEdgeWave, tpb, 0, stream>>>(Hbuf, src, dst, dinv, out);
}